// BiLSTMCRF_44006234914816
// MI455X (gfx1250) — compile-verified
//
#include <hip/hip_runtime.h>

typedef float v2f __attribute__((ext_vector_type(2)));
typedef float v4f __attribute__((ext_vector_type(4)));
typedef float v8f __attribute__((ext_vector_type(8)));

#define LOG2E 1.44269504088896340736f
#define LN2   0.69314718055994530942f

static constexpr int Bn = 2048, Ln = 512, Tn = 32;

// device-only intrinsics; host pass needs parseable stubs
__device__ __forceinline__ float fexp2(float x) {
#if defined(__HIP_DEVICE_COMPILE__)
  return __builtin_amdgcn_exp2f(x);
#else
  return x;
#endif
}
__device__ __forceinline__ float flog2(float x) {
#if defined(__HIP_DEVICE_COMPILE__)
  return __builtin_amdgcn_logf(x);
#else
  return x;
#endif
}
__device__ __forceinline__ float swap16(float x) {  // SWAPX16 across wave32 lane halves
#if defined(__HIP_DEVICE_COMPILE__)
  return __int_as_float(__builtin_amdgcn_ds_swizzle(__float_as_int(x), 0x401f));
#else
  return x;
#endif
}
__device__ __forceinline__ void sched_fence() {
#if defined(__HIP_DEVICE_COMPILE__)
  __builtin_amdgcn_sched_barrier(0);
#endif
}

// ---------------- gold path score: total[b] = sum_i<len gold[b,i] ----------------
__global__ __launch_bounds__(256) void crf_gold_kernel(
    const float* __restrict__ emit, const float* __restrict__ trans,
    const int* __restrict__ tags, const int* __restrict__ lengths,
    float* __restrict__ total)
{
  const int lane = threadIdx.x & 31;
  const int b    = (int)((blockIdx.x * blockDim.x + threadIdx.x) >> 5);
  const int len  = lengths[b];
  const int*   tg = tags + (size_t)b * Ln;
  const float* em = emit + (size_t)b * (Ln * Tn);
  float s = 0.f;
  for (int i = lane; i < len; i += 32) {
    int t = tg[i];
    float v = em[i * Tn + t];
    if (i > 0) v += trans[tg[i - 1] * Tn + t];
    s += v;
  }
#pragma unroll
  for (int off = 16; off; off >>= 1) s += __shfl_xor(s, off, 32);
  if (lane == 0) total[b] = s;
}

// ---------------- forward scan with WMMA f32 16x16x4, log2-domain ----------------
// Wave owns 16 batch rows. State d2 (= d * log2e) in WMMA C-layout:
//   d0[v] (tile0): tag k = v + 8*hl,   batch = wv*16 + (lane&15)
//   d1[v] (tile1): tag k = 16 + v + 8*hl
// C-layout regs are reused verbatim as next step's B-matrix; the layout
// permutation nu() is folded into the constant A = exp(trans)^T.
__global__ __launch_bounds__(256) void crf_scan_kernel(
    const float* __restrict__ emit, const float* __restrict__ trans,
    const int* __restrict__ lengths, const float* __restrict__ total,
    float* __restrict__ out)
{
  const int lane = threadIdx.x & 31;
  const int hl   = lane >> 4;        // lane-half
  const int ln16 = lane & 15;
  const int wv   = (int)((blockIdx.x * blockDim.x + threadIdx.x) >> 5);
  const int row  = wv * 16 + ln16;

  // Constant A-matrix: A[m][kpos] = exp(trans[nu(kpos)][m]) = 2^(trans*log2e).
  // A-layout (16x4 f32 chunk c, reg r): lane holds M=ln16, K = 4c + r + 2*hl.
  v2f A0[8], A1[8];
#pragma unroll
  for (int c = 0; c < 8; ++c) {
    v2f a0, a1;
#pragma unroll
    for (int r = 0; r < 2; ++r) {
      const int kpos = 4 * c + r + 2 * hl;
      const int rho  = 2 * (kpos >> 2) + (kpos & 1);          // C-state reg feeding this pos
      const int j    = (rho & 7) + 8 * ((kpos >> 1) & 1) + 16 * (rho >> 3); // tag at that reg
      a0[r] = fexp2(trans[j * Tn + (ln16 + 0)]  * LOG2E);
      a1[r] = fexp2(trans[j * Tn + (ln16 + 16)] * LOG2E);
    }
    A0[c] = a0; A1[c] = a1;
  }

  const int len = lengths[row];
  int maxlen = len;                       // wave-uniform loop bound
#pragma unroll
  for (int off = 16; off; off >>= 1) maxlen = max(maxlen, __shfl_xor(maxlen, off, 32));

  const float* ep = emit + (size_t)row * (Ln * Tn) + hl * 8;

  float d0[8], d1[8];
  { // d2(0) = emit[:,0,:] * log2e
    v4f e00 = *(const v4f*)(ep + 0);
    v4f e01 = *(const v4f*)(ep + 4);
    v4f e10 = *(const v4f*)(ep + 16);
    v4f e11 = *(const v4f*)(ep + 20);
#pragma unroll
    for (int v = 0; v < 4; ++v) {
      d0[v] = e00[v] * LOG2E; d0[v + 4] = e01[v] * LOG2E;
      d1[v] = e10[v] * LOG2E; d1[v + 4] = e11[v] * LOG2E;
    }
  }

  // double-buffered emission prefetch: n* always holds step-i data at loop entry
  v4f n00 = *(const v4f*)(ep + Tn + 0);
  v4f n01 = *(const v4f*)(ep + Tn + 4);
  v4f n10 = *(const v4f*)(ep + Tn + 16);
  v4f n11 = *(const v4f*)(ep + Tn + 20);

  for (int i = 1; i < maxlen; ++i) {
    const v4f e00 = n00, e01 = n01, e10 = n10, e11 = n11;
    {
      const int ip1 = (i + 1 < Ln) ? (i + 1) : (Ln - 1);
      const float* epn = ep + ip1 * Tn;
      n00 = *(const v4f*)(epn + 0);
      n01 = *(const v4f*)(epn + 4);
      n10 = *(const v4f*)(epn + 16);
      n11 = *(const v4f*)(epn + 20);
      const int ipf = (i + 8 < Ln) ? (i + 8) : (Ln - 1);
      __builtin_prefetch(ep + ipf * Tn, 0, 0);   // global_prefetch far-ahead
    }

    // per-batch row max over all 32 tags (16 local + cross-half swap)
    float mx = d0[0];
#pragma unroll
    for (int v = 1; v < 8; ++v) mx = fmaxf(mx, d0[v]);
#pragma unroll
    for (int v = 0; v < 8; ++v) mx = fmaxf(mx, d1[v]);
    mx = fmaxf(mx, swap16(mx));

    // p = 2^(d2 - mx), still in C-layout; pack straight into B-matrix chunks
    v2f bb[8];
#pragma unroll
    for (int c = 0; c < 8; ++c) {
      const int r0 = 2 * c, r1 = 2 * c + 1;
      v2f b;
      b[0] = fexp2(((r0 < 8) ? d0[r0] : d1[r0 - 8]) - mx);
      b[1] = fexp2(((r1 < 8) ? d0[r1] : d1[r1 - 8]) - mx);
      bb[c] = b;
    }
    sched_fence();  // all 16 exps issued before the WMMA burst

    // S = A x P : 4 independent chains of 4 chained K=4 WMMAs
    v8f a0lo = {}, a0hi = {}, a1lo = {}, a1hi = {};
#pragma unroll
    for (int c = 0; c < 4; ++c) {
      a0lo = __builtin_amdgcn_wmma_f32_16x16x4_f32(false, A0[c],     false, bb[c],
                                                   (short)0, a0lo, false, false);
      a1lo = __builtin_amdgcn_wmma_f32_16x16x4_f32(false, A1[c],     false, bb[c],
                                                   (short)0, a1lo, false, false);
      a0hi = __builtin_amdgcn_wmma_f32_16x16x4_f32(false, A0[c + 4], false, bb[c + 4],
                                                   (short)0, a0hi, false, false);
      a1hi = __builtin_amdgcn_wmma_f32_16x16x4_f32(false, A1[c + 4], false, bb[c + 4],
                                                   (short)0, a1hi, false, false);
    }
    sched_fence();
    const v8f acc0 = a0lo + a0hi;
    const v8f acc1 = a1lo + a1hi;

    // new_d2 = mx + log2(S) + e*log2e
    // branchless masked update: d += updf * (nd - d)  (no saveexec cascades)
    const float updf = (i < len) ? 1.0f : 0.0f;
#pragma unroll
    for (int v = 0; v < 8; ++v) {
      const float e0 = (v < 4) ? e00[v] : e01[v - 4];
      const float e1 = (v < 4) ? e10[v] : e11[v - 4];
      const float nd0 = fmaf(e0, LOG2E, flog2(acc0[v]) + mx);
      const float nd1 = fmaf(e1, LOG2E, flog2(acc1[v]) + mx);
      d0[v] = fmaf(updf, nd0 - d0[v], d0[v]);
      d1[v] = fmaf(updf, nd1 - d1[v], d1[v]);
    }
  }

  // log_z = ln2 * log2sumexp2 over 32 tags
  float mx = d0[0];
#pragma unroll
  for (int v = 1; v < 8; ++v) mx = fmaxf(mx, d0[v]);
#pragma unroll
  for (int v = 0; v < 8; ++v) mx = fmaxf(mx, d1[v]);
  mx = fmaxf(mx, swap16(mx));
  float s = 0.f;
#pragma unroll
  for (int v = 0; v < 8; ++v)
    s += fexp2(d0[v] - mx) + fexp2(d1[v] - mx);
  s += swap16(s);
  const float logz = LN2 * (flog2(s) + mx);

  if (hl == 0) out[row] = logz - total[row];
}

extern "C" void kernel_launch(void* const* d_in, const int* in_sizes, int n_in,
                              void* d_out, int out_size, void* d_ws, size_t ws_size,
                              hipStream_t stream) {
  const float* emit    = (const float*)d_in[0];
  const float* trans   = (const float*)d_in[1];
  const int*   tags    = (const int*)d_in[2];
  const int*   lengths = (const int*)d_in[3];
  float* out   = (float*)d_out;
  float* total = (float*)d_ws;   // 2048 floats of scratch

  // 1) gold path scores (also warms L2 with emit cachelines; 128MB fits in 192MB L2)
  crf_gold_kernel<<<Bn / 8, 256, 0, stream>>>(emit, trans, tags, lengths, total);
  // 2) WMMA forward scan: 2048/16 = 128 waves, 8 waves/block
  crf_scan_kernel<<<Bn / 128, 256, 0, stream>>>(emit, trans, lengths, total, out);
}